// AStarPathfinder_7662221656533
// MI455X (gfx1250) — compile-verified
//
#include <hip/hip_runtime.h>
#include <math.h>

#define GH 256
#define GW 256
#define GC 128
#define NPIX (GH * GW)
#define BIGF 1000000000.0f
#define KITERS 256
#define RAD 4                 // fused Jacobi sub-steps per launch
#define TILE 32
#define TIN (TILE + 2 * RAD)  // 40

typedef __attribute__((ext_vector_type(2))) float v2f;
typedef __attribute__((ext_vector_type(8))) float v8f;

__device__ __forceinline__ float waveReduceSum(float v) {
    // wave32 full reduction
    v += __shfl_xor(v, 16, 32);
    v += __shfl_xor(v, 8, 32);
    v += __shfl_xor(v, 4, 32);
    v += __shfl_xor(v, 2, 32);
    v += __shfl_xor(v, 1, 32);
    return v;
}

// ---------------------------------------------------------------------------
// 1) per-pixel 1/max(||f||,1e-12)   (one wave = one pixel, lane = 4 channels)
// ---------------------------------------------------------------------------
__global__ __launch_bounds__(256) void k_invnorm(const float* __restrict__ feat,
                                                 float* __restrict__ invn) {
    int wid = (blockIdx.x * blockDim.x + threadIdx.x) >> 5;
    int lane = threadIdx.x & 31;
    if (wid >= NPIX) return;
    float4 f = ((const float4*)feat)[wid * 32 + lane];
    float ss = f.x * f.x + f.y * f.y + f.z * f.z + f.w * f.w;
    ss = waveReduceSum(ss);
    if (lane == 0) invn[wid] = 1.0f / fmaxf(sqrtf(ss), 1e-12f);
}

// ---------------------------------------------------------------------------
// 2) cost grid: out[..][1+i] = 1 - fn[p].fn[p+dir_i], BIG at borders
// ---------------------------------------------------------------------------
__global__ __launch_bounds__(256) void k_cost(const float* __restrict__ feat,
                                              const float* __restrict__ invn,
                                              float* __restrict__ out) {
    int wid = (blockIdx.x * blockDim.x + threadIdx.x) >> 5;
    int lane = threadIdx.x & 31;
    if (wid >= NPIX) return;
    int y = wid / GW, x = wid % GW;
    float4 fp = ((const float4*)feat)[wid * 32 + lane];
    float ip = invn[wid];
    const int dxs[8] = {-1, -1, -1, 0, 0, 1, 1, 1};
    const int dys[8] = {-1, 0, 1, -1, 1, -1, 0, 1};
#pragma unroll
    for (int i = 0; i < 8; ++i) {
        int ny = y + dxs[i], nx = x + dys[i];
        float c = BIGF;
        if (ny >= 0 && ny < GH && nx >= 0 && nx < GW) {  // uniform per wave
            int q = ny * GW + nx;
            float4 fq = ((const float4*)feat)[q * 32 + lane];
            float d = fp.x * fq.x + fp.y * fq.y + fp.z * fq.z + fp.w * fq.w;
            d = waveReduceSum(d);
            c = 1.0f - d * ip * invn[q];
        }
        if (lane == 0) out[wid * 10 + 1 + i] = c;
    }
}

// ---------------------------------------------------------------------------
// 3) fused 3x3 pass: Sobel grad-mag mean (all 128 ch), local variance sum
//    (hf channels 64..127 -> lanes 16..31), ||lf - lf_end|| (lanes 0..15)
// ---------------------------------------------------------------------------
__global__ __launch_bounds__(256) void k_geovarabs(const float* __restrict__ feat,
                                                   const int* __restrict__ end_node,
                                                   float* __restrict__ geo,
                                                   float* __restrict__ var,
                                                   float* __restrict__ absb) {
    int wid = (blockIdx.x * blockDim.x + threadIdx.x) >> 5;
    int lane = threadIdx.x & 31;
    if (wid >= NPIX) return;
    int y = wid / GW, x = wid % GW;

    const float wxs[9] = {-1.f, 0.f, 1.f, -2.f, 0.f, 2.f, -1.f, 0.f, 1.f};
    const float wys[9] = {-1.f, -2.f, -1.f, 0.f, 0.f, 0.f, 1.f, 2.f, 1.f};

    float gx[4] = {0.f, 0.f, 0.f, 0.f}, gy[4] = {0.f, 0.f, 0.f, 0.f};
    float s1[4] = {0.f, 0.f, 0.f, 0.f}, s2[4] = {0.f, 0.f, 0.f, 0.f};
    float ct[4] = {0.f, 0.f, 0.f, 0.f};

#pragma unroll
    for (int k = 0; k < 9; ++k) {
        int ny = y + (k / 3) - 1, nx = x + (k % 3) - 1;
        if (ny < 0 || ny >= GH || nx < 0 || nx >= GW) continue;  // zero pad
        float4 v4 = ((const float4*)feat)[(ny * GW + nx) * 32 + lane];
        float v[4] = {v4.x, v4.y, v4.z, v4.w};
#pragma unroll
        for (int c = 0; c < 4; ++c) {
            gx[c] = fmaf(wxs[k], v[c], gx[c]);
            gy[c] = fmaf(wys[k], v[c], gy[c]);
            s1[c] += v[c];
            s2[c] = fmaf(v[c], v[c], s2[c]);
        }
        if (k == 4) { ct[0] = v[0]; ct[1] = v[1]; ct[2] = v[2]; ct[3] = v[3]; }
    }

    // geodesic: mean over channels of sqrt(gx^2+gy^2)
    float gm = 0.f;
#pragma unroll
    for (int c = 0; c < 4; ++c) gm += sqrtf(gx[c] * gx[c] + gy[c] * gy[c]);
    gm = waveReduceSum(gm) * (1.0f / 128.0f);

    // variance over hf channels: sum(avg(h^2) - avg(h)^2)
    float vv = 0.f;
#pragma unroll
    for (int c = 0; c < 4; ++c) {
        float mu = s1[c] * (1.0f / 9.0f);
        vv += s2[c] * (1.0f / 9.0f) - mu * mu;
    }
    vv = (lane >= 16) ? vv : 0.f;
    vv = waveReduceSum(vv);

    // absorption: ||lf - lf(end)||
    int ey = end_node[0], ex = end_node[1];
    float4 e4 = ((const float4*)feat)[(ey * GW + ex) * 32 + lane];
    float ev[4] = {e4.x, e4.y, e4.z, e4.w};
    float aa = 0.f;
#pragma unroll
    for (int c = 0; c < 4; ++c) {
        float d = ct[c] - ev[c];
        aa = fmaf(d, d, aa);
    }
    aa = (lane < 16) ? aa : 0.f;
    aa = sqrtf(waveReduceSum(aa));

    if (lane == 0) { geo[wid] = gm; var[wid] = vv; absb[wid] = aa; }
}

// ---------------------------------------------------------------------------
// 4) MLP via native FP32 WMMA: h = relu(X@W1+b1), omega = sigmoid(h@w2+b2)
//    one wave per 16 pixels; A 16x4 f32 (v2f), B 4x16 f32 (v2f), C 16x16 (v8f)
// ---------------------------------------------------------------------------
__global__ __launch_bounds__(32) void k_mlp(const float* __restrict__ feat,
                                            const float* __restrict__ w1,
                                            const float* __restrict__ b1,
                                            const float* __restrict__ w2,
                                            const float* __restrict__ b2,
                                            float* __restrict__ omega) {
    __shared__ float hsm[16][32];
    int lane = threadIdx.x;
    int half = lane >> 4;   // 0: K0/K1 rows, 1: K2/K3 rows
    int l15 = lane & 15;
    int r0 = blockIdx.x * 16;

    v8f c0 = {};
    v8f c1 = {};
#pragma unroll 4
    for (int t = 0; t < 32; ++t) {
        int kb = 4 * t + 2 * half;
        // A fragment: lane<16 holds row l15 K={kb,kb+1}; lane>=16 K={4t+2,4t+3}
        const float* xr = feat + (r0 + l15) * GC + kb;
        v2f a; a.x = xr[0]; a.y = xr[1];
        // B fragments for N-tiles 0..15 and 16..31 (w1 is row-major 128x32)
        v2f bA, bB;
        bA.x = w1[kb * 32 + l15];
        bA.y = w1[(kb + 1) * 32 + l15];
        bB.x = w1[kb * 32 + 16 + l15];
        bB.y = w1[(kb + 1) * 32 + 16 + l15];
        c0 = __builtin_amdgcn_wmma_f32_16x16x4_f32(false, a, false, bA, (short)0, c0, false, false);
        c1 = __builtin_amdgcn_wmma_f32_16x16x4_f32(false, a, false, bB, (short)0, c1, false, false);
    }
    // C/D layout: VGPR r, lanes 0-15 -> M=r, lanes 16-31 -> M=r+8; N = l15 (+16 for c1)
#pragma unroll
    for (int r = 0; r < 8; ++r) {
        int m = r + 8 * half;
        hsm[m][l15]      = fmaxf(c0[r] + b1[l15], 0.f);
        hsm[m][16 + l15] = fmaxf(c1[r] + b1[16 + l15], 0.f);
    }
    __syncthreads();
    if (lane < 16) {
        float acc = b2[0];
#pragma unroll
        for (int n = 0; n < 32; ++n) acc = fmaf(hsm[lane][n], w2[n], acc);
        omega[r0 + lane] = 1.0f / (1.0f + expf(-acc));
    }
}

// ---------------------------------------------------------------------------
// 5) combine heuristic -> out channel 0
// ---------------------------------------------------------------------------
__global__ __launch_bounds__(256) void k_combine(const float* __restrict__ geo,
                                                 const float* __restrict__ var,
                                                 const float* __restrict__ absb,
                                                 const float* __restrict__ omega,
                                                 const float* __restrict__ pd,
                                                 const float* __restrict__ pg,
                                                 const float* __restrict__ pb,
                                                 const int* __restrict__ end_node,
                                                 float* __restrict__ out) {
    int p = blockIdx.x * blockDim.x + threadIdx.x;
    if (p >= NPIX) return;
    float d = log1pf(expf(pd[0]));
    float g = log1pf(expf(pg[0]));
    float b = log1pf(expf(pb[0]));
    float ve = var[end_node[0] * GW + end_node[1]];
    float om = omega[p];
    float h = d * geo[p] + om * g * (ve - var[p]) + (1.0f - om) * b * absb[p];
    out[p * 10 + 0] = fmaxf(h, 0.f);
}

// ---------------------------------------------------------------------------
// 6) distance map
// ---------------------------------------------------------------------------
__global__ __launch_bounds__(256) void k_dist_init(float* __restrict__ dist,
                                                   const int* __restrict__ start) {
    int p = blockIdx.x * blockDim.x + threadIdx.x;
    if (p >= NPIX) return;
    dist[p] = (p == start[0] * GW + start[1]) ? 0.0f : BIGF;
}

// Fused Jacobi relaxation: RAD (=4) exact global sweeps per launch.
// 32x32 output tile, 40x40 dist halo tile + 40x40x8 cost slab in LDS.
// Sub-step s updates half-width (RAD-1-s) around the tile; its sources
// (half-width RAD-s) are exactly the region valid from the previous sub-step.
__global__ __launch_bounds__(256) void k_relax4(const float* __restrict__ prev,
                                                const float* __restrict__ out,
                                                float* __restrict__ next) {
    __shared__ float ld[2][TIN * TIN];       // 2 * 6400 B
    __shared__ float lc[TIN * TIN * 8];      // 51200 B
    const int tid = threadIdx.x;
    const int gy0 = blockIdx.y * TILE;
    const int gx0 = blockIdx.x * TILE;
    const int dxs[8] = {-1, -1, -1, 0, 0, 1, 1, 1};
    const int dys[8] = {-1, 0, 1, -1, 1, -1, 0, 1};

    // load dist halo tile + cost slab (BIG outside the global grid)
    for (int idx = tid; idx < TIN * TIN; idx += 256) {
        int ty = idx / TIN, tx = idx % TIN;
        int gy = gy0 - RAD + ty, gx = gx0 - RAD + tx;
        bool in = (gy >= 0 && gy < GH && gx >= 0 && gx < GW);
        int gp = gy * GW + gx;
        ld[0][idx] = in ? prev[gp] : BIGF;
#pragma unroll
        for (int i = 0; i < 8; ++i)
            lc[idx * 8 + i] = in ? out[gp * 10 + 1 + i] : BIGF;
    }

    int cur = 0;
#pragma unroll
    for (int s = 0; s < RAD; ++s) {
        __syncthreads();
        const int hw = (RAD - 1) - s;          // 3,2,1,0
        const int n = TILE + 2 * hw;           // 38,36,34,32
        const int off = RAD - hw;              // 1,2,3,4
        for (int idx = tid; idx < n * n; idx += 256) {
            int ry = idx / n, rx = idx % n;
            int ty = off + ry, tx = off + rx;
            int p = ty * TIN + tx;
            float d = ld[cur][p];
#pragma unroll
            for (int i = 0; i < 8; ++i) {
                int sp = (ty - dxs[i]) * TIN + (tx - dys[i]);
                d = fminf(d, ld[cur][sp] + lc[sp * 8 + i]);
            }
            ld[cur ^ 1][p] = d;
        }
        cur ^= 1;
    }
    __syncthreads();

    // write 32x32 interior
    for (int idx = tid; idx < TILE * TILE; idx += 256) {
        int ry = idx / TILE, rx = idx % TILE;
        int p = (RAD + ry) * TIN + (RAD + rx);
        next[(gy0 + ry) * GW + (gx0 + rx)] = fminf(ld[cur][p], BIGF);
    }
}

__global__ __launch_bounds__(256) void k_dist_out(const float* __restrict__ dist,
                                                  float* __restrict__ out) {
    int p = blockIdx.x * blockDim.x + threadIdx.x;
    if (p >= NPIX) return;
    out[p * 10 + 9] = dist[p];
}

// ---------------------------------------------------------------------------
extern "C" void kernel_launch(void* const* d_in, const int* in_sizes, int n_in,
                              void* d_out, int out_size, void* d_ws, size_t ws_size,
                              hipStream_t stream) {
    const float* feat  = (const float*)d_in[0];
    const float* delta = (const float*)d_in[1];
    const float* gamma = (const float*)d_in[2];
    const float* beta  = (const float*)d_in[3];
    const float* w1    = (const float*)d_in[4];
    const float* b1    = (const float*)d_in[5];
    const float* w2    = (const float*)d_in[6];
    const float* b2    = (const float*)d_in[7];
    const int*   start = (const int*)d_in[8];
    const int*   endn  = (const int*)d_in[9];
    float* out = (float*)d_out;

    float* ws    = (float*)d_ws;
    float* invn  = ws;                // 65536
    float* geo   = invn + NPIX;       // 65536
    float* var   = geo + NPIX;        // 65536
    float* absb  = var + NPIX;        // 65536
    float* omega = absb + NPIX;       // 65536
    float* distA = omega + NPIX;      // 65536
    float* distB = distA + NPIX;      // 65536

    const int waveBlocks = NPIX * 32 / 256;  // 8192 blocks of 8 waves
    const int pixBlocks  = NPIX / 256;       // 256 blocks

    k_invnorm<<<waveBlocks, 256, 0, stream>>>(feat, invn);
    k_cost<<<waveBlocks, 256, 0, stream>>>(feat, invn, out);
    k_geovarabs<<<waveBlocks, 256, 0, stream>>>(feat, endn, geo, var, absb);
    k_mlp<<<NPIX / 16, 32, 0, stream>>>(feat, w1, b1, w2, b2, omega);
    k_combine<<<pixBlocks, 256, 0, stream>>>(geo, var, absb, omega, delta, gamma, beta, endn, out);

    k_dist_init<<<pixBlocks, 256, 0, stream>>>(distA, start);
    dim3 tgrid(GW / TILE, GH / TILE);  // 8x8 tiles
    float* cur = distA;
    float* nxt = distB;
    for (int it = 0; it < KITERS / RAD; ++it) {   // 64 fused launches
        k_relax4<<<tgrid, 256, 0, stream>>>(cur, out, nxt);
        float* t = cur; cur = nxt; nxt = t;
    }
    k_dist_out<<<pixBlocks, 256, 0, stream>>>(cur, out);
}